// KnowledgeBaseLookup_4329327034923
// MI455X (gfx1250) — compile-verified
//
#include <hip/hip_runtime.h>
#include <cstdint>

// Problem shape (fixed by the reference): B=4, T=2048, K=16, C=262144, D=256.
namespace {
constexpr int kB = 4, kT = 2048, kK = 16, kC = 262144, kD = 256;
constexpr int kTokens = kB * kT;            // 8192
constexpr int kRowBytes = kD * 4;           // 1024 B per knowledge-base row
constexpr int kRowsPerWave = 2;             // 8 waves x 2 rows = 16 rows per token
}

typedef int   v4i __attribute__((ext_vector_type(4)));
typedef int   v8i __attribute__((ext_vector_type(8)));
typedef float v2f __attribute__((ext_vector_type(2)));
typedef float v8f __attribute__((ext_vector_type(8)));

// Issue one TDM gather (32-bit row indices) of kRowsPerWave rows of the
// (C x 256) f32 tensor into LDS at byte offset lds_byte_off.  Descriptor
// layout per CDNA5 ISA (D# groups 0..3, gather mode).
__device__ __forceinline__ void tdm_gather_rows(unsigned lds_byte_off,
                                                uint64_t kb_addr,
                                                int idx0, int idx1) {
  v4i g0;
  g0.x = (int)0xC0000001u;            // count=1 | 32-bit indices | gather_mode=1
  g0.y = (int)lds_byte_off;           // lds_addr (bytes, workgroup-relative)
  g0.z = (int)(uint32_t)(kb_addr & 0xFFFFFFFFu);                          // global_addr lo
  g0.w = (int)((uint32_t)((kb_addr >> 32) & 0x01FFFFFFu) | 0x80000000u);  // hi | type=2

  v8i g1;
  g1[0] = 0x00020000;                 // workgroup_mask=0, data_size=2 (4 bytes)
  g1[1] = (int)((kD & 0xFFFF) << 16); // tensor_dim0 = 256 (bits 79:48, low half)
  g1[2] = 0;                          // tensor_dim0 hi16 / tensor_dim1 lo16 (C lo16 = 0)
  g1[3] = (int)(((kD & 0xFFFF) << 16) | ((kC >> 16) & 0xFFFF)); // tile_dim0=256 | tensor_dim1 hi16
  g1[4] = kRowsPerWave;               // tile_dim1 = #valid gather indices
  g1[5] = kD;                         // tensor_dim0_stride = 256 elements
  g1[6] = 0;
  g1[7] = 0;

  v4i g2; g2.x = idx0; g2.y = idx1; g2.z = 0; g2.w = 0;   // row_index_0..1 (32-bit mode)
  v4i g3; g3.x = 0;    g3.y = 0;    g3.z = 0; g3.w = 0;

  asm volatile("tensor_load_to_lds %0, %1, %2, %3"
               :: "s"(g0), "s"(g1), "s"(g2), "s"(g3)
               : "memory");
}

__global__ __launch_bounds__(kD)
void knowledge_gather_tdm_kernel(const float* __restrict__ weights,   // (B,T,K) f32
                                 const int*   __restrict__ indexes,   // (B,T,K) int64 as dword pairs
                                 const float* __restrict__ kb,        // (C,D) f32
                                 float*       __restrict__ out) {     // (B,T,D) f32
  __shared__ float smem[kK * kD];     // 16 gathered rows, 16 KB

  const int token = blockIdx.x;
  const float* wp = weights + (size_t)token * kK;
  const int*   ip = indexes + (size_t)token * kK * 2;   // int64 -> low dword at ip[2*r]

  // Uniform weight loads (scalar path).
  float wv[kK];
#pragma unroll
  for (int r = 0; r < kK; ++r) wv[r] = wp[r];

  // --- TDM gather: every wave unconditionally issues exactly ONE gather of
  // its 2 rows (TDM ignores EXEC, so no divergent guard may wrap it). ---
  const int wave = threadIdx.x >> 5;                 // 0..7
  const int lane = threadIdx.x & 31;
  const int r0   = wave * kRowsPerWave;              // rows r0, r0+1
  const int i0   = ip[2 * r0];                       // low dword of int64 index
  const int i1   = ip[2 * (r0 + 1)];
  const unsigned lds_base = (unsigned)(uintptr_t)(&smem[0]); // low 32 bits = LDS offset
  tdm_gather_rows(lds_base + (unsigned)(r0 * kRowBytes),
                  (uint64_t)(uintptr_t)kb, i0, i1);
  __builtin_amdgcn_s_wait_tensorcnt(0);              // this wave's rows are in LDS
  __syncthreads();                                   // all 16 rows visible to all waves

#if __has_builtin(__builtin_amdgcn_wmma_f32_16x16x4_f32)
  // --- Weighted K-reduction on the matrix pipe (free: kernel is HBM-bound).
  // out_chunk(1x16) = w(1x16) x G_chunk(16x16) via 4 chained 16x16x4 WMMAs.
  // A(16x4): weights replicated down M.  Documented f32 A layout:
  //   VGPR0 = w[k0 + (lane<16 ? 0 : 2)],  VGPR1 = w[k0 + (lane<16 ? 1 : 3)]
  // B(4x16): mirrored layout, rows from LDS:
  //   VGPR0 lanes0-15 = row k0+0, lanes16-31 = row k0+2 (cols cb..cb+15)
  //   VGPR1 lanes0-15 = row k0+1, lanes16-31 = row k0+3
  // Every M-row of D equals the chunk result; M=0 lives in lanes 0-15 of
  // acc[0], the identical M=8 row in lanes 16-31 -> one coalesced store.
  const int  n   = lane & 15;
  const bool hi  = lane >= 16;
  const int  cb0 = wave * 32;          // chunk 0 columns
  const int  cb1 = wave * 32 + 16;     // chunk 1 columns
  const int  rA  = hi ? 2 : 0;         // B VGPR0 k-offset for this half-wave
  const int  rB  = hi ? 3 : 1;         // B VGPR1 k-offset

  v8f acc0 = {};
  v8f acc1 = {};
#pragma unroll
  for (int p = 0; p < 4; ++p) {
    const int k0 = p * 4;
    v2f a;
    a.x = hi ? wv[k0 + 2] : wv[k0 + 0];
    a.y = hi ? wv[k0 + 3] : wv[k0 + 1];
    v2f b0, b1;
    b0.x = smem[(k0 + rA) * kD + cb0 + n];
    b0.y = smem[(k0 + rB) * kD + cb0 + n];
    b1.x = smem[(k0 + rA) * kD + cb1 + n];
    b1.y = smem[(k0 + rB) * kD + cb1 + n];
    acc0 = __builtin_amdgcn_wmma_f32_16x16x4_f32(false, a, false, b0,
                                                 (short)0, acc0, false, false);
    acc1 = __builtin_amdgcn_wmma_f32_16x16x4_f32(false, a, false, b1,
                                                 (short)0, acc1, false, false);
  }
  // lanes 0-15: chunk0 col cb0+n ; lanes 16-31: chunk1 col cb1+n == wave*32+lane
  const float val = hi ? acc1[0] : acc0[0];
  out[(size_t)token * kD + wave * 32 + lane] = val;
#else
  // Fallback: per-thread scalar FMA reduction (thread = output column).
  const int col = threadIdx.x;
  float acc = 0.0f;
#pragma unroll
  for (int r = 0; r < kK; ++r)
    acc = __builtin_fmaf(wv[r], smem[r * kD + col], acc);
  out[(size_t)token * kD + col] = acc;
#endif
}

extern "C" void kernel_launch(void* const* d_in, const int* in_sizes, int n_in,
                              void* d_out, int out_size, void* d_ws, size_t ws_size,
                              hipStream_t stream) {
  const float* weights = (const float*)d_in[0];
  const int*   indexes = (const int*)d_in[1];   // int64 payload, addressed as dword pairs
  const float* kb      = (const float*)d_in[2];
  float*       out     = (float*)d_out;

  dim3 grid(kTokens);
  dim3 block(kD);
  hipLaunchKernelGGL(knowledge_gather_tdm_kernel, grid, block, 0, stream,
                     weights, indexes, kb, out);
}